// model004_37666863186445
// MI455X (gfx1250) — compile-verified
//
#include <hip/hip_runtime.h>

typedef __attribute__((ext_vector_type(2))) float v2f;
typedef __attribute__((ext_vector_type(8))) float v8f;

namespace {
constexpr int kT      = 4096;
constexpr int kNB     = 64;
constexpr int kNU     = 32;
constexpr int kNY     = 32;
constexpr int kNH     = 256;
constexpr int kNHalf  = 128;
constexpr int kChunk  = 64;                // timesteps per chunk
constexpr int kNChunk = kT / kChunk;       // 64 chunks
constexpr int kBG     = 32;                // batches per block
constexpr int kNBG    = kNB / kBG;         // 2 batch groups
constexpr int kTG     = 2;                 // timesteps per inner group
constexpr int kNGrp   = kChunk / kTG;      // 32 groups per chunk

// workspace layout (float offsets)
constexpr size_t WS_LR  = 0;
constexpr size_t WS_LI  = WS_LR + kNHalf;
constexpr size_t WS_LRL = WS_LI + kNHalf;
constexpr size_t WS_LIL = WS_LRL + kNHalf;
constexpr size_t WS_BP  = WS_LIL + kNHalf;                    // 256*32 (nf-scaled B)
constexpr size_t WS_X0  = WS_BP + (size_t)kNH * kNU;          // 64*256
constexpr size_t WS_F   = WS_X0 + (size_t)kNB * kNH;          // 64*64*256 chunk-local finals
constexpr size_t WS_S   = WS_F + (size_t)kNChunk * kNB * kNH; // 64*64*256 chunk-initial states

constexpr size_t kSmemBytes = (size_t)kTG * kBG * kNH * sizeof(float); // 65536
} // namespace

// ---------------------------------------------------------------------------
// Setup: rotation params (lr, li), R^64 via repeated squaring, nf-scaled B,
// and x0 = y0 @ y2x_w^T + y2x_b.
// ---------------------------------------------------------------------------
__global__ void __launch_bounds__(256)
setup_kernel(const float* __restrict__ y0, const float* __restrict__ llr,
             const float* __restrict__ lli, const float* __restrict__ Bmat,
             const float* __restrict__ y2xw, const float* __restrict__ y2xb,
             float* __restrict__ ws) {
  __shared__ float s_nf[kNH];
  const int tid = threadIdx.x;
  if (tid < kNHalf) {
    const float decay = __expf(-fabsf(llr[tid]));
    const float r     = __expf(-decay);
    const float th    = 1.57079632679489662f * __expf(-fabsf(lli[tid]));
    const float lr    = r * __cosf(th);
    const float li    = r * __sinf(th);
    const float nfh   = sqrtf(fmaxf(0.f, 1.f - r * r));
    ws[WS_LR + tid] = lr;
    ws[WS_LI + tid] = li;
    // (lr + i*li)^64 via 6 complex squarings
    float pr = lr, pi = li;
    #pragma unroll
    for (int s = 0; s < 6; ++s) {
      const float nr = pr * pr - pi * pi;
      const float ni = 2.f * pr * pi;
      pr = nr; pi = ni;
    }
    ws[WS_LRL + tid] = pr;
    ws[WS_LIL + tid] = pi;
    s_nf[tid]          = nfh;
    s_nf[tid + kNHalf] = nfh;
  }
  __syncthreads();
  // Bp[n][u] = B[n][u] * nf[n]   (fold Bu scaling into the weight)
  for (int i = tid; i < kNH * kNU; i += blockDim.x) {
    ws[WS_BP + i] = Bmat[i] * s_nf[i / kNU];
  }
  // x0[b][n] = y0[b] . y2x_w[n] + y2x_b[n]
  for (int o = tid; o < kNB * kNH; o += blockDim.x) {
    const int b = o >> 8;
    const int n = o & 255;
    float acc = y2xb[n];
    const float* yr = y0 + b * kNY;
    const float* wr = y2xw + n * kNY;
    #pragma unroll 8
    for (int u = 0; u < kNY; ++u) acc += yr[u] * wr[u];
    ws[WS_X0 + o] = acc;
  }
}

// ---------------------------------------------------------------------------
// Bu GEMM for one timestep slot: (32b x 32u) x (32u x 256n) via f32 WMMA.
// Each wave: fixed (slot, mt), loops 8 n-tiles, 8 k-steps each.
// Writes D-fragments into the LDS X buffer.
// ---------------------------------------------------------------------------
__device__ __forceinline__ void bu_gemm_store(const float* __restrict__ Uslot,
                                              const float* __restrict__ Bp,
                                              float* __restrict__ Xslot,
                                              int wave, int laneM, int laneHi) {
  const int mt    = (wave >> 1) & 1;
  const int ntPar = wave & 1;
  const int m     = mt * 16 + laneM;
  const int koff  = laneHi << 1;
  const float* Arow = Uslot + m * kNU;
  #pragma unroll
  for (int j = 0; j < 8; ++j) {
    const int nt = ntPar + (j << 1);
    const int n  = nt * 16 + laneM;
    const float* Brow = Bp + n * kNU;
    v8f acc = {};
    #pragma unroll
    for (int kk = 0; kk < 8; ++kk) {
      const int k = (kk << 2) + koff;
      v2f a = *(const v2f*)(Arow + k);   // A[m][k], A[m][k+1]
      v2f b = *(const v2f*)(Brow + k);   // B[k][n], B[k+1][n] == Bp[n][k..k+1]
      acc = __builtin_amdgcn_wmma_f32_16x16x4_f32(false, a, false, b,
                                                  (short)0, acc, false, false);
    }
    #pragma unroll
    for (int v = 0; v < 8; ++v) {
      const int mm = mt * 16 + v + (laneHi << 3);
      Xslot[mm * kNH + n] = acc[v];
    }
  }
}

// ---------------------------------------------------------------------------
// Pass 1: per (chunk, batch-group) block, run local scan from zero state,
// keep state in registers, emit chunk-final state F[c].
// ---------------------------------------------------------------------------
__global__ void __launch_bounds__(256)
pass1_kernel(const float* __restrict__ U, float* __restrict__ ws) {
  extern __shared__ float Xbuf[];   // [kTG][kBG][kNH]
  const float* Bp = ws + WS_BP;

  const int g = blockIdx.x;     // batch group
  const int c = blockIdx.y;     // chunk
  const int tid    = threadIdx.x;
  const int lane   = tid & 31;
  const int wave   = tid >> 5;
  const int laneM  = lane & 15;
  const int laneHi = lane >> 4;
  const int slotW  = wave >> 2;

  const int sb = tid >> 3;            // local batch owned by this thread
  const int h0 = (tid & 7) << 4;      // first complex pair index

  float st1[16], st2[16], rlr[16], rli[16];
  #pragma unroll
  for (int i = 0; i < 16; ++i) {
    st1[i] = 0.f; st2[i] = 0.f;
    rlr[i] = ws[WS_LR + h0 + i];
    rli[i] = ws[WS_LI + h0 + i];
  }

  for (int grp = 0; grp < kNGrp; ++grp) {
    const int t0 = c * kChunk + grp * kTG;
    const float* Uslot = U + ((size_t)(t0 + slotW) * kNB + g * kBG) * kNU;
    bu_gemm_store(Uslot, Bp, Xbuf + slotW * (kBG * kNH), wave, laneM, laneHi);
    __syncthreads();
    #pragma unroll
    for (int s = 0; s < kTG; ++s) {
      const float* Bu = Xbuf + s * (kBG * kNH) + sb * kNH;
      #pragma unroll
      for (int i = 0; i < 16; ++i) {
        const float x1 = st1[i], x2 = st2[i];
        st1[i] = rlr[i] * x1 - rli[i] * x2 + Bu[h0 + i];
        st2[i] = rli[i] * x1 + rlr[i] * x2 + Bu[h0 + i + kNHalf];
      }
    }
    __syncthreads();
  }
  float* F = ws + WS_F + ((size_t)c * kNB + g * kBG + sb) * kNH;
  #pragma unroll
  for (int i = 0; i < 16; ++i) {
    F[h0 + i]          = st1[i];
    F[h0 + i + kNHalf] = st2[i];
  }
}

// ---------------------------------------------------------------------------
// Combine: S[0] = x0 ; S[c+1] = R^64 * S[c] + F[c].  One block, register state.
// ---------------------------------------------------------------------------
__global__ void __launch_bounds__(256)
combine_kernel(float* __restrict__ ws) {
  const int tid = threadIdx.x;
  for (int i = 0; i < 32; ++i) {         // 8192 pairs / 256 threads
    const int p = tid * 32 + i;
    const int b = p >> 7;
    const int h = p & 127;
    float s1 = ws[WS_X0 + (size_t)b * kNH + h];
    float s2 = ws[WS_X0 + (size_t)b * kNH + h + kNHalf];
    const float pr = ws[WS_LRL + h];
    const float pi = ws[WS_LIL + h];
    for (int c = 0; c < kNChunk; ++c) {
      const size_t base = (size_t)c * kNB * kNH + (size_t)b * kNH + h;
      ws[WS_S + base]          = s1;
      ws[WS_S + base + kNHalf] = s2;
      const float f1 = ws[WS_F + base];
      const float f2 = ws[WS_F + base + kNHalf];
      const float n1 = pr * s1 - pi * s2 + f1;
      const float n2 = pi * s1 + pr * s2 + f2;
      s1 = n1; s2 = n2;
    }
  }
}

// ---------------------------------------------------------------------------
// Pass 2: replay each chunk from S[c]; fuse Bu GEMM -> scan (in-place in LDS)
// -> Y projection GEMM (WMMA, K=256) -> store Y. X never touches HBM.
// ---------------------------------------------------------------------------
__global__ void __launch_bounds__(256)
pass2_kernel(const float* __restrict__ U, const float* __restrict__ W,
             const float* __restrict__ bias, const float* __restrict__ ws,
             float* __restrict__ out) {
  extern __shared__ float Xbuf[];   // [kTG][kBG][kNH]
  const float* Bp = ws + WS_BP;

  const int g = blockIdx.x;
  const int c = blockIdx.y;
  const int tid    = threadIdx.x;
  const int lane   = tid & 31;
  const int wave   = tid >> 5;
  const int laneM  = lane & 15;
  const int laneHi = lane >> 4;
  const int koff   = laneHi << 1;
  const int slotW  = wave >> 2;

  const int sb = tid >> 3;
  const int h0 = (tid & 7) << 4;

  float st1[16], st2[16], rlr[16], rli[16];
  {
    const float* S = ws + WS_S + ((size_t)c * kNB + g * kBG + sb) * kNH;
    #pragma unroll
    for (int i = 0; i < 16; ++i) {
      st1[i] = S[h0 + i];
      st2[i] = S[h0 + i + kNHalf];
      rlr[i] = ws[WS_LR + h0 + i];
      rli[i] = ws[WS_LI + h0 + i];
    }
  }

  // Y-GEMM tile for this wave: one 16x16 output tile per timestep slot
  const int mt2 = (wave >> 1) & 1;
  const int nt2 = wave & 1;
  const float* Wrow  = W + (size_t)(nt2 * 16 + laneM) * kNH;  // x2y_w[y][.]
  const float  biasl = bias[nt2 * 16 + laneM];

  for (int grp = 0; grp < kNGrp; ++grp) {
    const int t0 = c * kChunk + grp * kTG;
    const float* Uslot = U + ((size_t)(t0 + slotW) * kNB + g * kBG) * kNU;
    bu_gemm_store(Uslot, Bp, Xbuf + slotW * (kBG * kNH), wave, laneM, laneHi);
    __syncthreads();
    // scan with in-place write-back: Bu slot becomes X_t
    #pragma unroll
    for (int s = 0; s < kTG; ++s) {
      float* Bu = Xbuf + s * (kBG * kNH) + sb * kNH;
      #pragma unroll
      for (int i = 0; i < 16; ++i) {
        const float x1 = st1[i], x2 = st2[i];
        const float n1 = rlr[i] * x1 - rli[i] * x2 + Bu[h0 + i];
        const float n2 = rli[i] * x1 + rlr[i] * x2 + Bu[h0 + i + kNHalf];
        st1[i] = n1; st2[i] = n2;
        Bu[h0 + i]          = n1;
        Bu[h0 + i + kNHalf] = n2;
      }
    }
    __syncthreads();
    // Y_t = X_t @ x2y_w^T + x2y_b   (16x16 tile per wave, K = 256)
    {
      const float* Arow = Xbuf + slotW * (kBG * kNH) + (mt2 * 16 + laneM) * kNH;
      v8f acc = {};
      #pragma unroll 8
      for (int kk = 0; kk < 64; ++kk) {
        const int k = (kk << 2) + koff;
        v2f a = *(const v2f*)(Arow + k);
        v2f b = *(const v2f*)(Wrow + k);
        acc = __builtin_amdgcn_wmma_f32_16x16x4_f32(false, a, false, b,
                                                    (short)0, acc, false, false);
      }
      const int t    = t0 + slotW;
      const int ycol = nt2 * 16 + laneM;
      float* Orow = out + (size_t)t * kNB * kNY + (size_t)(g * kBG) * kNY;
      #pragma unroll
      for (int v = 0; v < 8; ++v) {
        const int mm = mt2 * 16 + v + (laneHi << 3);
        Orow[mm * kNY + ycol] = acc[v] + biasl;
      }
    }
    __syncthreads();
  }
}

// ---------------------------------------------------------------------------
extern "C" void kernel_launch(void* const* d_in, const int* in_sizes, int n_in,
                              void* d_out, int out_size, void* d_ws, size_t ws_size,
                              hipStream_t stream) {
  (void)in_sizes; (void)n_in; (void)out_size; (void)ws_size;
  const float* y0   = (const float*)d_in[0];
  const float* U    = (const float*)d_in[1];
  const float* llr  = (const float*)d_in[2];
  const float* lli  = (const float*)d_in[3];
  const float* Bm   = (const float*)d_in[4];
  const float* y2xw = (const float*)d_in[5];
  const float* y2xb = (const float*)d_in[6];
  const float* x2yw = (const float*)d_in[7];
  const float* x2yb = (const float*)d_in[8];
  float* ws  = (float*)d_ws;
  float* out = (float*)d_out;

  hipFuncSetAttribute(reinterpret_cast<const void*>(pass1_kernel),
                      hipFuncAttributeMaxDynamicSharedMemorySize, (int)kSmemBytes);
  hipFuncSetAttribute(reinterpret_cast<const void*>(pass2_kernel),
                      hipFuncAttributeMaxDynamicSharedMemorySize, (int)kSmemBytes);

  setup_kernel<<<1, 256, 0, stream>>>(y0, llr, lli, Bm, y2xw, y2xb, ws);
  dim3 grid(kNBG, kNChunk);
  pass1_kernel<<<grid, 256, kSmemBytes, stream>>>(U, ws);
  combine_kernel<<<1, 256, 0, stream>>>(ws);
  pass2_kernel<<<grid, 256, kSmemBytes, stream>>>(U, x2yw, x2yb, ws, out);
}